// V32Model_43250320670986
// MI455X (gfx1250) — compile-verified
//
#include <hip/hip_runtime.h>
#include <hip/hip_bf16.h>
#include <math.h>

#define S_LEN   512
#define D_MODEL 768
#define DIN     1536
#define H_HEADS 24
#define N_STATE 64
#define HEADDIM 64
#define DCONV   4
#define CONV_CH (DIN + 2*N_STATE)             // 1664
#define DIP     (2*DIN + 2*N_STATE + H_HEADS) // 3224 (true width)
#define DIP_P   3232                          // padded to multiple of 16
#define R_LORA  8
#define NL0     6
#define NL1     6
#define MAXLOOP 8
#define V_VOCAB 50432
#define EPSV    1e-5f

typedef __attribute__((ext_vector_type(2))) float v2f;
typedef __attribute__((ext_vector_type(4))) float v4f;
typedef __attribute__((ext_vector_type(8))) float v8f;

__device__ __forceinline__ float sigm(float x) { return 1.f / (1.f + expf(-x)); }

__device__ __forceinline__ v8f wmma4(v2f a, v2f b, v8f c) {
    return __builtin_amdgcn_wmma_f32_16x16x4_f32(false, a, false, b, (short)0, c, false, false);
}
__device__ __forceinline__ v2f lo2(v4f v) { v2f r; r.x = v.x; r.y = v.y; return r; }
__device__ __forceinline__ v2f hi2(v4f v) { v2f r; r.x = v.z; r.y = v.w; return r; }

// ---------------------------------------------------------------------------
// GEMM: C[M,N] = A[M,K] * B[N,K]^T, f32 row-major, V_WMMA_F32_16X16X4_F32.
// REQUIRES: M, N multiples of 16; K multiple of 16 (caller pads).
// One 16x16 tile per wave, 8 waves/block. Branch-free software-pipelined
// inner loop: per 16 K-values each lane does two b128 loads of A and two of
// B (4 contiguous floats at row[k + 4*(lane>>4)]), feeding 4 WMMAs via a
// K-permutation applied identically to A and B (K sum order is irrelevant):
//   wmma#0 <- K {k..k+1 (lanes<16), k+4..k+5 (lanes>=16)}  = q.xy
//   wmma#1 <- K {k+2..k+3,          k+6..k+7}              = q.zw
// Loads of chunk i+1 issue before the WMMAs of chunk i; final chunk peeled.
// Store rows are clamped to Mst (caller may zero-pad M up to 16).
// ---------------------------------------------------------------------------
__launch_bounds__(256)
__global__ void gemm_f32_wmma(const float* __restrict__ A, const float* __restrict__ B,
                              float* __restrict__ C, int M, int N, int K, int Mst)
{
    const int lane = threadIdx.x;   // 0..31
    const int wv   = threadIdx.y;   // 0..7
    const int tilesN = N >> 4;
    const int tilesM = M >> 4;
    const int id = blockIdx.x * 8 + wv;
    if (id >= tilesM * tilesN) return;
    const int tm = id / tilesN;
    const int tn = id % tilesN;

    const int l16 = lane & 15;
    const int q4  = (lane >> 4) << 2;   // 0 or 4
    const float* __restrict__ arow = A + (size_t)(tm * 16 + l16) * K + q4;
    const float* __restrict__ brow = B + (size_t)(tn * 16 + l16) * K + q4;

    v8f acc0 = {0.f,0.f,0.f,0.f,0.f,0.f,0.f,0.f};
    v8f acc1 = acc0, acc2 = acc0, acc3 = acc0;

    // prologue: load chunk 0
    v4f a0 = *(const v4f*)(arow);
    v4f a1 = *(const v4f*)(arow + 8);
    v4f b0 = *(const v4f*)(brow);
    v4f b1 = *(const v4f*)(brow + 8);
    // steady state: branch-free; loads of chunk i+1 ahead of WMMAs of chunk i
    for (int k = 16; k + 16 <= K; k += 16) {
        const v4f na0 = *(const v4f*)(arow + k);
        const v4f na1 = *(const v4f*)(arow + k + 8);
        const v4f nb0 = *(const v4f*)(brow + k);
        const v4f nb1 = *(const v4f*)(brow + k + 8);
        acc0 = wmma4(lo2(a0), lo2(b0), acc0);
        acc1 = wmma4(hi2(a0), hi2(b0), acc1);
        acc2 = wmma4(lo2(a1), lo2(b1), acc2);
        acc3 = wmma4(hi2(a1), hi2(b1), acc3);
        a0 = na0; a1 = na1; b0 = nb0; b1 = nb1;
    }
    // epilogue: consume final chunk
    acc0 = wmma4(lo2(a0), lo2(b0), acc0);
    acc1 = wmma4(hi2(a0), hi2(b0), acc1);
    acc2 = wmma4(lo2(a1), lo2(b1), acc2);
    acc3 = wmma4(hi2(a1), hi2(b1), acc3);

    acc0 = (acc0 + acc1) + (acc2 + acc3);

    const int cn  = tn * 16 + l16;
    const int cm0 = tm * 16 + ((lane >> 4) << 3);
    #pragma unroll
    for (int r = 0; r < 8; ++r) {
        const int cm = cm0 + r;
        if (cm < Mst) C[(size_t)cm * N + cn] = acc0[r];
    }
}

// ---------------------------------------------------------------------------
// Elementwise / small kernels
// ---------------------------------------------------------------------------
__global__ void zero_fill(float* __restrict__ p, int n)
{
    const int idx = blockIdx.x * blockDim.x + threadIdx.x;
    if (idx < n) p[idx] = 0.f;
}

__global__ void copy_mat(const float* __restrict__ in, float* __restrict__ out, int n)
{
    const int idx = blockIdx.x * blockDim.x + threadIdx.x;
    if (idx < n) out[idx] = in[idx];
}

__global__ void embed_gather(const float* __restrict__ emb, const int* __restrict__ ids,
                             float* __restrict__ x)
{
    const int idx = blockIdx.x * blockDim.x + threadIdx.x;
    if (idx >= S_LEN * D_MODEL) return;
    const int s = idx / D_MODEL, d = idx % D_MODEL;
    x[idx] = emb[(size_t)ids[s] * D_MODEL + d];
}

// W_eff[e,d] = W[e,d] + 2 * sum_r Bm[e,r] * Am[r,d]
__global__ void lora_eff(const float* __restrict__ W, const float* __restrict__ Bm,
                         const float* __restrict__ Am, float* __restrict__ out,
                         int E, int Dd)
{
    const int idx = blockIdx.x * blockDim.x + threadIdx.x;
    if (idx >= E * Dd) return;
    const int e = idx / Dd, d = idx % Dd;
    float acc = W[idx];
    #pragma unroll
    for (int r = 0; r < R_LORA; ++r)
        acc += 2.f * Bm[e * R_LORA + r] * Am[r * Dd + d];
    out[idx] = acc;
}

// res = (addRes ? res + x : x); xn = rmsnorm(res) * w   (one block per row s)
__launch_bounds__(256)
__global__ void resadd_rmsnorm(float* __restrict__ res, const float* __restrict__ x,
                               const float* __restrict__ w, float* __restrict__ out,
                               int addRes)
{
    __shared__ float red[256];
    const int s = blockIdx.x, tid = threadIdx.x;
    float v[3]; float ss = 0.f;
    #pragma unroll
    for (int i = 0; i < 3; ++i) {
        const int d = tid + i * 256;
        float r = x[(size_t)s * D_MODEL + d];
        if (addRes) r += res[(size_t)s * D_MODEL + d];
        res[(size_t)s * D_MODEL + d] = r;
        v[i] = r; ss += r * r;
    }
    red[tid] = ss; __syncthreads();
    for (int o = 128; o > 0; o >>= 1) { if (tid < o) red[tid] += red[tid + o]; __syncthreads(); }
    const float scale = rsqrtf(red[0] / (float)D_MODEL + EPSV);
    #pragma unroll
    for (int i = 0; i < 3; ++i) {
        const int d = tid + i * 256;
        out[(size_t)s * D_MODEL + d] = v[i] * scale * w[d];
    }
}

__launch_bounds__(256)
__global__ void rmsnorm_inplace(float* __restrict__ x, const float* __restrict__ w)
{
    __shared__ float red[256];
    const int s = blockIdx.x, tid = threadIdx.x;
    float v[3]; float ss = 0.f;
    #pragma unroll
    for (int i = 0; i < 3; ++i) {
        const int d = tid + i * 256;
        v[i] = x[(size_t)s * D_MODEL + d]; ss += v[i] * v[i];
    }
    red[tid] = ss; __syncthreads();
    for (int o = 128; o > 0; o >>= 1) { if (tid < o) red[tid] += red[tid + o]; __syncthreads(); }
    const float scale = rsqrtf(red[0] / (float)D_MODEL + EPSV);
    #pragma unroll
    for (int i = 0; i < 3; ++i) {
        const int d = tid + i * 256;
        x[(size_t)s * D_MODEL + d] = v[i] * scale * w[d];
    }
}

// dt = softplus(zx[:, 3200+h] + dtb);  dA = exp(-exp(Alog)*dt)   (zx stride DIP_P)
__global__ void dtprep(const float* __restrict__ zx, const float* __restrict__ dtb,
                       const float* __restrict__ Alog, float* __restrict__ dt,
                       float* __restrict__ dA)
{
    const int idx = blockIdx.x * blockDim.x + threadIdx.x;
    if (idx >= S_LEN * H_HEADS) return;
    const int s = idx / H_HEADS, h = idx % H_HEADS;
    const float raw = zx[(size_t)s * DIP_P + DIN + CONV_CH + h] + dtb[h];
    const float sp = (raw > 20.f) ? raw : log1pf(expf(raw));
    dt[idx] = sp;
    dA[idx] = expf(-expf(Alog[h]) * sp);
}

// causal depthwise conv (DCONV=4) over zx cols [DIN, DIN+CONV_CH) + silu
__global__ void conv_silu(const float* __restrict__ zx, const float* __restrict__ cw,
                          const float* __restrict__ cb, float* __restrict__ xc)
{
    const int idx = blockIdx.x * blockDim.x + threadIdx.x;
    if (idx >= S_LEN * CONV_CH) return;
    const int s = idx / CONV_CH, c = idx % CONV_CH;
    float acc = cb[c];
    #pragma unroll
    for (int kk = 0; kk < DCONV; ++kk) {
        const int sp = s - (DCONV - 1) + kk;
        if (sp >= 0) acc += zx[(size_t)sp * DIP_P + DIN + c] * cw[c * DCONV + kk];
    }
    xc[idx] = acc * sigm(acc);
}

// Sequential SSM scan. grid = H heads; block 256. State h[p=64][n=64] in regs
// (16 per thread). thread t: p = t>>2, n-range = (t&3)*16..+16.
__launch_bounds__(256)
__global__ void scan_kernel(const float* __restrict__ xc, const float* __restrict__ dtv,
                            const float* __restrict__ dav, const float* __restrict__ Dh,
                            float* __restrict__ y)
{
    __shared__ float xr[HEADDIM];
    __shared__ float Br[N_STATE];
    __shared__ float Cr[N_STATE];
    __shared__ float ypart[256];
    const int hh = blockIdx.x;
    const int t  = threadIdx.x;
    const int p  = t >> 2;
    const int nq = t & 3;
    const float Dval = Dh[hh];
    float hs[16];
    #pragma unroll
    for (int i = 0; i < 16; ++i) hs[i] = 0.f;

    for (int s = 0; s < S_LEN; ++s) {
        const float* row = xc + (size_t)s * CONV_CH;
        if (t < HEADDIM)                       xr[t]                       = row[hh * HEADDIM + t];
        else if (t < HEADDIM + N_STATE)        Br[t - HEADDIM]             = row[DIN + (t - HEADDIM)];
        else if (t < HEADDIM + 2 * N_STATE)    Cr[t - HEADDIM - N_STATE]   = row[DIN + N_STATE + (t - HEADDIM - N_STATE)];
        __syncthreads();
        const float da   = dav[s * H_HEADS + hh];
        const float dtt  = dtv[s * H_HEADS + hh];
        const float coef = dtt * xr[p];
        float ps = 0.f;
        #pragma unroll
        for (int i = 0; i < 16; ++i) {
            const int n = nq * 16 + i;
            hs[i] = hs[i] * da + coef * Br[n];
            ps += hs[i] * Cr[n];
        }
        ypart[t] = ps;
        __syncthreads();
        if (t < HEADDIM) {
            const float yy = ypart[4 * t] + ypart[4 * t + 1] + ypart[4 * t + 2] + ypart[4 * t + 3];
            y[(size_t)s * DIN + hh * HEADDIM + t] = yy + Dval * xr[t];
        }
    }
}

// yn = rmsnorm( y * silu(z) ) * gn_w   over DIN; z = zx[:, 0:DIN] (stride DIP_P)
__launch_bounds__(256)
__global__ void gate_rmsnorm(const float* __restrict__ y, const float* __restrict__ zx,
                             const float* __restrict__ gw, float* __restrict__ out)
{
    __shared__ float red[256];
    const int s = blockIdx.x, tid = threadIdx.x;
    float g[6]; float ss = 0.f;
    #pragma unroll
    for (int i = 0; i < 6; ++i) {
        const int d = tid + i * 256;
        const float z  = zx[(size_t)s * DIP_P + d];
        const float gv = y[(size_t)s * DIN + d] * (z * sigm(z));
        g[i] = gv; ss += gv * gv;
    }
    red[tid] = ss; __syncthreads();
    for (int o = 128; o > 0; o >>= 1) { if (tid < o) red[tid] += red[tid + o]; __syncthreads(); }
    const float scale = rsqrtf(red[0] / (float)DIN + EPSV);
    #pragma unroll
    for (int i = 0; i < 6; ++i) {
        const int d = tid + i * 256;
        out[(size_t)s * DIN + d] = g[i] * scale * gw[d];
    }
}

__global__ void addvec(float* __restrict__ x, const float* __restrict__ t, int n)
{
    const int idx = blockIdx.x * blockDim.x + threadIdx.x;
    if (idx < n) x[idx] += t[idx];
}

// x += gate[0]*xpr + step_emb[d]
__global__ void loop_inject(float* __restrict__ x, const float* __restrict__ xpr,
                            const float* __restrict__ gate, const float* __restrict__ semb)
{
    const int idx = blockIdx.x * blockDim.x + threadIdx.x;
    if (idx >= S_LEN * D_MODEL) return;
    x[idx] += gate[0] * xpr[idx] + semb[idx % D_MODEL];
}

// h = rmsnorm(x[S-1] + res[S-1]) * norm_f_w  -> out (D)
__launch_bounds__(256)
__global__ void final_h(const float* __restrict__ x, const float* __restrict__ res,
                        const float* __restrict__ w, float* __restrict__ out)
{
    __shared__ float red[256];
    const int tid = threadIdx.x;
    const size_t base = (size_t)(S_LEN - 1) * D_MODEL;
    float v[3]; float ss = 0.f;
    #pragma unroll
    for (int i = 0; i < 3; ++i) {
        const int d = tid + i * 256;
        v[i] = x[base + d] + res[base + d]; ss += v[i] * v[i];
    }
    red[tid] = ss; __syncthreads();
    for (int o = 128; o > 0; o >>= 1) { if (tid < o) red[tid] += red[tid + o]; __syncthreads(); }
    const float scale = rsqrtf(red[0] / (float)D_MODEL + EPSV);
    #pragma unroll
    for (int i = 0; i < 3; ++i) {
        const int d = tid + i * 256;
        out[d] = v[i] * scale * w[d];
    }
}

// ---------------------------------------------------------------------------
// Host orchestration
// ---------------------------------------------------------------------------
extern "C" void kernel_launch(void* const* d_in, const int* in_sizes, int n_in,
                              void* d_out, int out_size, void* d_ws, size_t ws_size,
                              hipStream_t stream)
{
    (void)in_sizes; (void)n_in; (void)out_size; (void)ws_size;

    const float* emb        = (const float*)d_in[0];
    const float* lm_head_W  = (const float*)d_in[1];
    const float* norm_f_w   = (const float*)d_in[2];
    const float* step_emb_w = (const float*)d_in[3];
    const float* loop_norm_w= (const float*)d_in[4];
    const float* lifeline   = (const float*)d_in[5];
    const float* base_ln_w  = (const float*)d_in[6];
    const float* base_inW   = (const float*)d_in[7];
    const float* base_convw = (const float*)d_in[8];
    const float* base_convb = (const float*)d_in[9];
    const float* base_Alog  = (const float*)d_in[10];
    const float* base_D     = (const float*)d_in[11];
    const float* base_dtb   = (const float*)d_in[12];
    const float* base_gn_w  = (const float*)d_in[13];
    const float* base_outW  = (const float*)d_in[14];
    const float* lp_ln_w    = (const float*)d_in[15];
    const float* lp_inW     = (const float*)d_in[16];
    const float* lp_inA     = (const float*)d_in[17];
    const float* lp_inB     = (const float*)d_in[18];
    const float* lp_convw   = (const float*)d_in[19];
    const float* lp_convb   = (const float*)d_in[20];
    const float* lp_Alog    = (const float*)d_in[21];
    const float* lp_D       = (const float*)d_in[22];
    const float* lp_dtb     = (const float*)d_in[23];
    const float* lp_gn_w    = (const float*)d_in[24];
    const float* lp_outW    = (const float*)d_in[25];
    const float* lp_outA    = (const float*)d_in[26];
    const float* lp_outB    = (const float*)d_in[27];
    const float* core_inW   = (const float*)d_in[28];
    const float* core_convw = (const float*)d_in[29];
    const float* core_convb = (const float*)d_in[30];
    const float* core_Alog  = (const float*)d_in[31];
    const float* core_D     = (const float*)d_in[32];
    const float* core_dtb   = (const float*)d_in[33];
    const float* core_gn_w  = (const float*)d_in[34];
    const float* core_outW  = (const float*)d_in[35];
    const int*   ids        = (const int*)d_in[36];
    float* out = (float*)d_out;

    // workspace carve-up (~37 MB)
    float* p = (float*)d_ws;
    auto carve = [&](size_t n) { float* r = p; p += n; return r; };
    float* inW_eff  = carve((size_t)DIP_P * D_MODEL);   // 3232 rows: 3224 live + 8 zero
    float* outW_eff = carve((size_t)D_MODEL * DIN);
    float* x    = carve((size_t)S_LEN * D_MODEL);
    float* res  = carve((size_t)S_LEN * D_MODEL);
    float* xn   = carve((size_t)S_LEN * D_MODEL);
    float* xpr  = carve((size_t)S_LEN * D_MODEL);
    float* zx   = carve((size_t)S_LEN * DIP_P);         // row stride DIP_P
    float* xc   = carve((size_t)S_LEN * CONV_CH);
    float* dtbf = carve((size_t)S_LEN * H_HEADS);
    float* dabf = carve((size_t)S_LEN * H_HEADS);
    float* yb   = carve((size_t)S_LEN * DIN);
    float* yn   = carve((size_t)S_LEN * DIN);
    float* hb   = carve((size_t)16 * D_MODEL);          // 8 live rows + 8 zero rows

    auto gemm = [&](const float* A, const float* B, float* C, int M, int N, int K, int Mst) {
        const int tiles = (M >> 4) * (N >> 4);
        const int blocks = (tiles + 7) / 8;
        gemm_f32_wmma<<<dim3(blocks), dim3(32, 8), 0, stream>>>(A, B, C, M, N, K, Mst);
    };

    // mamba2 mixer; inW must point at the padded inW_eff buffer (DIP_P rows)
    auto mamba = [&](const float* xin, float* xout, const float* inWp, const float* cw,
                     const float* cb, const float* Alog, const float* Dh,
                     const float* dtbias, const float* gnw, const float* oW) {
        gemm(xin, inWp, zx, S_LEN, DIP_P, D_MODEL, S_LEN);
        dtprep<<<(S_LEN * H_HEADS + 255) / 256, 256, 0, stream>>>(zx, dtbias, Alog, dtbf, dabf);
        conv_silu<<<(S_LEN * CONV_CH + 255) / 256, 256, 0, stream>>>(zx, cw, cb, xc);
        scan_kernel<<<H_HEADS, 256, 0, stream>>>(xc, dtbf, dabf, Dh, yb);
        gate_rmsnorm<<<S_LEN, 256, 0, stream>>>(yb, zx, gnw, yn);
        gemm(yn, oW, xout, S_LEN, D_MODEL, DIN, S_LEN);
    };

    auto blockfn = [&](int addRes, const float* lnw, const float* inWp, const float* cw,
                       const float* cb, const float* Alog, const float* Dh,
                       const float* dtbias, const float* gnw, const float* oW) {
        resadd_rmsnorm<<<S_LEN, 256, 0, stream>>>(res, x, lnw, xn, addRes);
        mamba(xn, x, inWp, cw, cb, Alog, Dh, dtbias, gnw, oW);
    };

    auto loopblock = [&](int l, int addRes) {
        lora_eff<<<((size_t)DIP * D_MODEL + 255) / 256, 256, 0, stream>>>(
            lp_inW + (size_t)l * DIP * D_MODEL, lp_inB + (size_t)l * DIP * R_LORA,
            lp_inA + (size_t)l * R_LORA * D_MODEL, inW_eff, DIP, D_MODEL);
        lora_eff<<<((size_t)D_MODEL * DIN + 255) / 256, 256, 0, stream>>>(
            lp_outW + (size_t)l * D_MODEL * DIN, lp_outB + (size_t)l * D_MODEL * R_LORA,
            lp_outA + (size_t)l * R_LORA * DIN, outW_eff, D_MODEL, DIN);
        blockfn(addRes, lp_ln_w + (size_t)l * D_MODEL, inW_eff,
                lp_convw + (size_t)l * CONV_CH * DCONV, lp_convb + (size_t)l * CONV_CH,
                lp_Alog + (size_t)l * H_HEADS, lp_D + (size_t)l * H_HEADS,
                lp_dtb + (size_t)l * H_HEADS, lp_gn_w + (size_t)l * DIN, outW_eff);
    };

    // ---- forward pass ----
    embed_gather<<<(S_LEN * D_MODEL + 255) / 256, 256, 0, stream>>>(emb, ids, x);
    // zero the pad rows once per call (never overwritten afterwards)
    zero_fill<<<(8 * D_MODEL + 255) / 256, 256, 0, stream>>>(
        inW_eff + (size_t)DIP * D_MODEL, (DIP_P - DIP) * D_MODEL);
    zero_fill<<<(8 * D_MODEL + 255) / 256, 256, 0, stream>>>(hb + 8 * D_MODEL, 8 * D_MODEL);

    for (int l = 0; l < NL0; ++l) {
        // stage this layer's inW into the padded buffer (pad rows stay zero)
        copy_mat<<<((size_t)DIP * D_MODEL + 255) / 256, 256, 0, stream>>>(
            base_inW + (size_t)l * DIP * D_MODEL, inW_eff, DIP * D_MODEL);
        blockfn(l > 0 ? 1 : 0, base_ln_w + (size_t)l * D_MODEL, inW_eff,
                base_convw + (size_t)l * CONV_CH * DCONV, base_convb + (size_t)l * CONV_CH,
                base_Alog + (size_t)l * H_HEADS, base_D + (size_t)l * H_HEADS,
                base_dtb + (size_t)l * H_HEADS, base_gn_w + (size_t)l * DIN,
                base_outW + (size_t)l * D_MODEL * DIN);
    }

    for (int l = 0; l < NL1; ++l) loopblock(l, 1);

    hipMemcpyAsync(xpr, x, (size_t)S_LEN * D_MODEL * sizeof(float),
                   hipMemcpyDeviceToDevice, stream);

    for (int li = 0; li < MAXLOOP; ++li) {
        loop_inject<<<(S_LEN * D_MODEL + 255) / 256, 256, 0, stream>>>(
            x, xpr, lifeline, step_emb_w + (size_t)li * D_MODEL);
        for (int l = 0; l < NL1; ++l) loopblock(l, 1);
        // x = x + mamba2(x, core...)
        copy_mat<<<((size_t)DIP * D_MODEL + 255) / 256, 256, 0, stream>>>(
            core_inW, inW_eff, DIP * D_MODEL);
        mamba(x, xn, inW_eff, core_convw, core_convb, core_Alog, core_D,
              core_dtb, core_gn_w, core_outW);
        addvec<<<(S_LEN * D_MODEL + 255) / 256, 256, 0, stream>>>(x, xn, S_LEN * D_MODEL);
        rmsnorm_inplace<<<S_LEN, 256, 0, stream>>>(x, loop_norm_w);
        final_h<<<1, 256, 0, stream>>>(x, res, norm_f_w, hb + (size_t)li * D_MODEL);
    }

    // logits: (8 x V) = hb(16-pad x 768) @ lm_head_W^T; store clamped to the
    // 8 real rows. One pass over the 155 MB vocab matrix.
    gemm(hb, lm_head_W, out, 16, V_VOCAB, D_MODEL, MAXLOOP);
}